// SelfCorrelationComputation_584115552551
// MI455X (gfx1250) — compile-verified
//
#include <hip/hip_runtime.h>

typedef __attribute__((ext_vector_type(2))) float v2f;
typedef __attribute__((ext_vector_type(4))) float v4f;
typedef __attribute__((ext_vector_type(8))) float v8f;

#define BATCH 16
#define CH    64
#define HDIM  56
#define WDIM  56
#define TILE  8
#define HALO  12            // TILE + 4 (5x5 window, pad 2)
#define NPOS  (HALO * HALO) // 144
#define PS    145           // odd LDS stride -> bank-conflict free
#define TAPS  25

__global__ __launch_bounds__(256)
void selfcorr_kernel(const float* __restrict__ x, float* __restrict__ out) {
    __shared__ float xs[CH * PS];   // relu(x) tile, later normalized in place
    __shared__ float invn[NPOS];    // per-position 1/max(||.||, eps)

    const int tid = threadIdx.x;
    const int tw  = blockIdx.x;     // 0..6
    const int th  = blockIdx.y;     // 0..6
    const int b   = blockIdx.z;     // 0..15
    const int h0  = th * TILE;
    const int w0  = tw * TILE;

    // ---- Phase 1: load relu(x) halo tile (each HBM input byte read once) ----
    for (int idx = tid; idx < CH * NPOS; idx += 256) {
        int c  = idx / NPOS;
        int p  = idx - c * NPOS;
        int hh = p / HALO;
        int ww = p - hh * HALO;
        int gh = h0 + hh - 2;
        int gw = w0 + ww - 2;
        float v = 0.0f;
        if (gh >= 0 && gh < HDIM && gw >= 0 && gw < WDIM) {
            v = fmaxf(x[((b * CH + c) * HDIM + gh) * WDIM + gw], 0.0f);
        }
        xs[c * PS + p] = v;
    }
    __syncthreads();

    // ---- Phase 2: channel sum-of-squares via V_WMMA_F32_16X16X4_F32 --------
    // D = A(16 pos x 4 ch of r^2) * B(ones) + C  =>  every column of D holds
    // the row sum; accumulate 16 WMMAs to cover all 64 channels.
    {
        const int wave = tid >> 5;
        const int lane = tid & 31;
        const int m    = lane & 15;          // A-matrix row (position)
        const int kb   = (lane >> 4) << 1;   // K base: lanes 0-15 -> K0,1; 16-31 -> K2,3
        v2f bones; bones.x = 1.0f; bones.y = 1.0f;

        for (int chunk = wave; chunk < 9; chunk += 8) {   // 9*16 = 144 positions
            const int pbase = chunk * 16;
            v8f acc = {};
            #pragma unroll
            for (int kc = 0; kc < 16; ++kc) {
                int c0 = kc * 4 + kb;
                float r0 = xs[c0 * PS + pbase + m];
                float r1 = xs[(c0 + 1) * PS + pbase + m];
                v2f a; a.x = r0 * r0; a.y = r1 * r1;
                acc = __builtin_amdgcn_wmma_f32_16x16x4_f32(
                    false, a, false, bones, (short)0, acc, false, false);
            }
            // D layout: VGPR v, lanes 0-15 -> M=v ; lanes 16-31 -> M=v+8.
            if ((lane & 15) == 0) {
                int mb = (lane >> 4) << 3;   // 0 or 8
                #pragma unroll
                for (int v = 0; v < 8; ++v) {
                    float s = acc[v];
                    invn[pbase + mb + v] = 1.0f / fmaxf(sqrtf(s), 1e-12f);
                }
            }
        }
    }
    __syncthreads();

    // ---- Phase 3: normalize tile in place ----------------------------------
    for (int idx = tid; idx < CH * NPOS; idx += 256) {
        int c = idx / NPOS;
        int p = idx - c * NPOS;
        xs[c * PS + p] *= invn[p];
    }
    __syncthreads();

    // ---- Phase 4: stream 64*8*8*25 outputs, float4 non-temporal stores -----
    // Output runs of 200 contiguous floats (8 w * 25 taps per (c,h) row);
    // 200 % 4 == 0 and bases are 4-aligned, so 4-chunks never straddle runs.
    const int total = CH * TILE * TILE * TAPS;  // 102400 = 1024 * 100
    for (int o = tid * 4; o < total; o += 1024) {
        v4f res;
        int gidx0 = 0;
        #pragma unroll
        for (int j = 0; j < 4; ++j) {
            int oo = o + j;
            int t  = oo % 25;
            int r  = oo / 25;          // c*64 + hl*8 + wl
            int wl = r & 7;
            int hl = (r >> 3) & 7;
            int c  = r >> 6;
            int di = t / 5;
            int dj = t - di * 5;
            float ctr = xs[c * PS + (hl + 2) * HALO + (wl + 2)];
            float nb  = xs[c * PS + (hl + di) * HALO + (wl + dj)];
            res[j] = ctr * nb;
            if (j == 0)
                gidx0 = (((b * CH + c) * HDIM + (h0 + hl)) * WDIM + (w0 + wl)) * 25 + t;
        }
        __builtin_nontemporal_store(res, (v4f*)(out + gidx0));
    }
}

extern "C" void kernel_launch(void* const* d_in, const int* in_sizes, int n_in,
                              void* d_out, int out_size, void* d_ws, size_t ws_size,
                              hipStream_t stream) {
    const float* x = (const float*)d_in[0];
    float* out = (float*)d_out;
    dim3 grid(7, 7, BATCH);   // (tiles_w, tiles_h, batch)
    dim3 block(256);
    selfcorr_kernel<<<grid, block, 0, stream>>>(x, out);
}